// attention_mask_54185307406440
// MI455X (gfx1250) — compile-verified
//
#include <hip/hip_runtime.h>
#include <math.h>

typedef __attribute__((ext_vector_type(2))) float v2f;
typedef __attribute__((ext_vector_type(4))) float v4f;
typedef __attribute__((ext_vector_type(8))) float v8f;

#define C_DIM 16
#define S_DIM 16384   // 128*128
#define TOPK  8192    // max(1, int(S * 0.5))

// lane L <-> lane L^16 swap: v_permlanex16_b32 with identity selects (pure VALU,
// avoids the ds_bpermute/LDS path __shfl_xor(.,16) lowers to).
__device__ __forceinline__ float swap16(float v) {
    int i = __builtin_bit_cast(int, v);
    int r = __builtin_amdgcn_permlanex16(i, i, 0x76543210, 0xfedcba98, false, false);
    return __builtin_bit_cast(float, r);
}

// ---------------------------------------------------------------------------
// Kernel 1: flash-attention style pass. Each wave owns 16 query rows.
//   simT_tile = K_tile x Q^T          (4x chained V_WMMA_F32_16X16X4_F32)
//   p = exp2(simT)  with 0.25*log2(e) pre-folded into Q (loaded once)
//   yT += V^T x p                     (4x chained WMMA, accumulator = C operand)
// At the end: upd[s] = sigmoid(max_c y[s][c] / denom[s])
// No running max: logits are bounded (|q.k|/4 has sigma~1), f32 exp is safe.
// ---------------------------------------------------------------------------
__global__ __launch_bounds__(256) void attn_upd_kernel(const float* __restrict__ ego,
                                                       const float* __restrict__ cav,
                                                       float* __restrict__ upd) {
    const int lane = threadIdx.x & 31;
    const int wave = blockIdx.x * (blockDim.x >> 5) + (threadIdx.x >> 5);
    const int s0   = wave * 16;            // first query row of this wave
    const int r    = lane & 15;            // row/col index within tile
    const bool hi  = lane >= 16;           // upper half-wave (K-chunk +2/+3)

    // B1 = Q^T in B-layout, pre-scaled by (1/sqrt(C)) * log2(e) so the per-tile
    // logits feed v_exp_f32 (2^x) directly with no extra multiplies.
    const float SC = 0.25f * 1.44269504088896340736f;
    v2f b1[4];
#pragma unroll
    for (int t = 0; t < 4; ++t) {
        const int kx = 4 * t + (hi ? 2 : 0);
        b1[t].x = ego[kx * S_DIM + s0 + r] * SC;
        b1[t].y = ego[(kx + 1) * S_DIM + s0 + r] * SC;
    }

    v8f   y     = {};     // y^T accumulator: lane L, vgpr i <-> y[s=L&15][c=i+8*hi]
    float denom = 0.0f;   // per-lane partial softmax denominator for query r

    for (int j = 0; j < S_DIM; j += 16) {
        // A1 = K tile in A-layout (A[m][c] = cav[c*S + j + m])
        v2f a1[4];
#pragma unroll
        for (int t = 0; t < 4; ++t) {
            const int kx = 4 * t + (hi ? 2 : 0);
            a1[t].x = cav[kx * S_DIM + j + r];
            a1[t].y = cav[(kx + 1) * S_DIM + j + r];
        }
        // A2 = V^T in A-layout (A[c][tk] = cav[c*S + j + tk]); contiguous b128 loads
        v2f a2[4];
#pragma unroll
        for (int t = 0; t < 4; ++t) {
            const v4f v = *(const v4f*)(cav + r * S_DIM + j + 4 * t);
            a2[t].x = hi ? v.z : v.x;
            a2[t].y = hi ? v.w : v.y;
        }
        __builtin_prefetch(cav + r * S_DIM + j + 16, 0, 3);   // next tile, keep near

        // GEMM1: simT tile (16 keys x 16 queries), K-dim = C = 16 via 4 chained WMMAs
        v8f s = {};
#pragma unroll
        for (int t = 0; t < 4; ++t)
            s = __builtin_amdgcn_wmma_f32_16x16x4_f32(false, a1[t], false, b1[t],
                                                      (short)0, s, false, false);

        // exponentiate (scale already folded into Q), accumulate denominator
        v8f   p;
        float part = 0.0f;
#pragma unroll
        for (int i = 0; i < 8; ++i) {
            p[i] = __builtin_amdgcn_exp2f(s[i]);   // v_exp_f32 = 2^x, no extra mul
            part += p[i];
        }
        denom += part;

        // Re-layout P (D layout, key-major) into B-layout for GEMM2:
        // B[k][n] lives at vgpr (k&7), lane (n + 16*(k>=8)) -> one half-wave swap.
        float ps[8];
#pragma unroll
        for (int i = 0; i < 8; ++i) ps[i] = swap16(p[i]);
        v2f b2[4];
        b2[0].x = hi ? ps[2] : p[0];  b2[0].y = hi ? ps[3] : p[1];   // k = 0..3
        b2[1].x = hi ? ps[6] : p[4];  b2[1].y = hi ? ps[7] : p[5];   // k = 4..7
        b2[2].x = hi ? p[2] : ps[0];  b2[2].y = hi ? p[3] : ps[1];   // k = 8..11
        b2[3].x = hi ? p[6] : ps[4];  b2[3].y = hi ? p[7] : ps[5];   // k = 12..15

        // GEMM2: yT += V^T x P  (accumulate through the WMMA C operand)
#pragma unroll
        for (int t = 0; t < 4; ++t)
            y = __builtin_amdgcn_wmma_f32_16x16x4_f32(false, a2[t], false, b2[t],
                                                      (short)0, y, false, false);
    }

    // finalize: denom over both half-waves, max over 16 channels, sigmoid
    const float dtot = denom + swap16(denom);
    float m = y[0];
#pragma unroll
    for (int i = 1; i < 8; ++i) m = fmaxf(m, y[i]);
    m = fmaxf(m, swap16(m));
    const float u = 1.0f / (1.0f + __expf(-m / dtot));
    if (!hi) upd[s0 + r] = u;
}

// ---------------------------------------------------------------------------
// Kernel 2: single-workgroup bitonic sort (descending) of 16384 upd values in
// LDS; threshold = k-th largest (index TOPK-1).
// ---------------------------------------------------------------------------
__global__ __launch_bounds__(1024) void select_kernel(const float* __restrict__ upd,
                                                      float* __restrict__ thr) {
    __shared__ float sd[S_DIM];
    for (int i = threadIdx.x; i < S_DIM; i += blockDim.x) sd[i] = upd[i];
    __syncthreads();
    for (int k = 2; k <= S_DIM; k <<= 1) {
        for (int j = k >> 1; j > 0; j >>= 1) {
            for (int i = threadIdx.x; i < S_DIM; i += blockDim.x) {
                const int ixj = i ^ j;
                if (ixj > i) {
                    const float a = sd[i], b = sd[ixj];
                    const bool desc = ((i & k) == 0);
                    if (desc ? (a < b) : (a > b)) { sd[i] = b; sd[ixj] = a; }
                }
            }
            __syncthreads();
        }
    }
    if (threadIdx.x == 0) thr[0] = sd[TOPK - 1];
}

// ---------------------------------------------------------------------------
// Kernel 3: binary mask
// ---------------------------------------------------------------------------
__global__ void mask_kernel(const float* __restrict__ upd,
                            const float* __restrict__ thr,
                            float* __restrict__ out) {
    const int i = blockIdx.x * blockDim.x + threadIdx.x;
    if (i < S_DIM) out[i] = (upd[i] >= thr[0]) ? 1.0f : 0.0f;
}

extern "C" void kernel_launch(void* const* d_in, const int* in_sizes, int n_in,
                              void* d_out, int out_size, void* d_ws, size_t ws_size,
                              hipStream_t stream) {
    const float* ego = (const float*)d_in[0];   // (16,128,128) f32
    const float* cav = (const float*)d_in[1];   // (16,128,128) f32
    float* upd = (float*)d_ws;                  // 16384 f32
    float* thr = upd + S_DIM;                   // 1 f32
    float* out = (float*)d_out;                 // 16384 f32 mask

    // 1024 waves total (one per 16-query block): 128 blocks x 8 waves
    attn_upd_kernel<<<128, 256, 0, stream>>>(ego, cav, upd);
    select_kernel<<<1, 1024, 0, stream>>>(upd, thr);
    mask_kernel<<<S_DIM / 256, 256, 0, stream>>>(upd, thr, out);
}